// YoloLoss_90640989815103
// MI455X (gfx1250) — compile-verified
//
#include <hip/hip_runtime.h>
#include <stdint.h>

// Problem constants (from reference)
#define SS     49          // 7*7 cells
#define CCH    30          // channels per cell
#define NBATCH 4096
#define NPAIR  (NBATCH * SS)      // 200704 (b,cell) rows
#define ROWF   (SS * CCH)         // floats per batch item (1470)

// Workspace layout (float indices)
#define WS_M   0                  // [2][49][32] masked max   (3136 floats)
#define WS_D   3136               // [2][49][32] denom        (3136 floats)
#define WS_ACC 6272               // scalar accumulator

typedef unsigned int u32x4 __attribute__((ext_vector_type(4)));
typedef int          i32x8 __attribute__((ext_vector_type(8)));
typedef int          i32x4 __attribute__((ext_vector_type(4)));

// ---------------- CDNA5 async tensor path helpers ----------------
// Low 32 bits of a flat pointer into LDS == LDS byte offset (per CDNA5
// aperture rules).
__device__ __forceinline__ unsigned lds_byte_off(const void* p) {
  return (unsigned)(uintptr_t)p;
}

// GLOBAL_LOAD_ASYNC_TO_LDS_B128: vdst = VGPR holding LDS byte address,
// vaddr = 64-bit global address, no saddr ("off"). Tracked by ASYNCcnt.
__device__ __forceinline__ void async_ld_b128(unsigned lds_byte, const void* g) {
  asm volatile("global_load_async_to_lds_b128 %0, %1, off"
               :: "v"(lds_byte), "v"((unsigned long long)(uintptr_t)g)
               : "memory");
}

__device__ __forceinline__ void wait_async0() {
  asm volatile("s_wait_asynccnt 0" ::: "memory");
}

// Tensor Data Mover: DMA a contiguous span of `bytes` (multiple of 8) from
// global memory into LDS. One instruction per call, TENSORcnt-tracked.
// D# packing per CDNA5 ISA ch.8 (2-D tensor: bytes/8 x 1 of 8-byte elements).
// This toolchain (clang-23 / therock) exposes the 6-arg builtin:
// (u32x4 g0, i32x8 g1, i32x4 g2, i32x4 g3, i32x8 extra, i32 cpol).
__device__ __forceinline__ void tdm_load_span(unsigned lds_byte,
                                              const void* g, unsigned bytes) {
  const unsigned long long ga = (unsigned long long)(uintptr_t)g;
  const unsigned n8 = bytes >> 3;            // elements of size 8 B
  u32x4 g0;
  g0[0] = 1u;                                 // count=1, user mode
  g0[1] = lds_byte;                           // lds_addr
  g0[2] = (unsigned)ga;                       // global_addr[31:0]
  g0[3] = (unsigned)(ga >> 32) | (2u << 30);  // global_addr[56:32] | type=2
  i32x8 g1;
  g1[0] = (3 << 16);                          // wg_mask=0, data_size=3 (8B)
  g1[1] = (int)(n8 << 16);                    // tensor_dim0[15:0]
  g1[2] = (1 << 16);                          // tensor_dim0[31:16]=0, tensor_dim1=1
  g1[3] = (int)(n8 << 16);                    // tensor_dim1 hi=0, tile_dim0
  g1[4] = 1;                                  // tile_dim1=1, tile_dim2=0
  g1[5] = (int)n8;                            // tensor_dim0_stride[31:0]
  g1[6] = 0;                                  // stride0 hi, stride1 lo
  g1[7] = 0;                                  // stride1 hi
  i32x4 gz4 = {0, 0, 0, 0};                   // groups 2/3: unused (2-D)
  i32x8 gz8 = {0, 0, 0, 0, 0, 0, 0, 0};
  __builtin_amdgcn_tensor_load_to_lds(g0, g1, gz4, gz4, gz8, 0);
}

// ---------------- init: M = -inf bits, D = 0, ACC = 0 ----------------
__global__ void k_init(float* __restrict__ ws) {
  int i = blockIdx.x * 256 + threadIdx.x;
  if (i < 3136) {
    ((int*)ws)[i] = 0xFF800000;           // -inf as int bits (for atomicMax)
  } else if (i <= 6272) {
    ws[i] = 0.0f;                          // denom + accumulator
  }
}

// ---------------- pass A: masked max over batch -----------------------
// grid: (8 batch chunks, 49 cells, 2 boxes), block: 256 (8 wave32)
__global__ __launch_bounds__(256)
void k_max(const float* __restrict__ y, const float* __restrict__ yh,
           float* __restrict__ ws) {
  const int t    = threadIdx.x;
  const int z    = blockIdx.x;       // batch chunk (512 items)
  const int cell = blockIdx.y;
  const int k    = blockIdx.z;
  const int nch  = 25 - 5 * k;
  const int cb   = 5 * k;

  float vmax[25];
#pragma unroll
  for (int c = 0; c < 25; ++c) vmax[c] = -__builtin_inff();

  const size_t cellOff = (size_t)cell * CCH;
#pragma unroll
  for (int i = 0; i < 2; ++i) {
    int b = z * 512 + i * 256 + t;
    size_t rowo = (size_t)b * ROWF + cellOff;
    if (i == 0) __builtin_prefetch(&y[rowo + (size_t)256 * ROWF], 0, 0);
    float conf = y[rowo + cb];
    if (conf == 1.0f) {
      const float* rh = yh + rowo + cb + 5;
#pragma unroll
      for (int c = 0; c < 25; ++c)
        if (c < nch) vmax[c] = fmaxf(vmax[c], rh[c]);
    }
  }
  // wave32 xor-reduction (all lanes end with the max)
#pragma unroll
  for (int c = 0; c < 25; ++c) {
#pragma unroll
    for (int off = 16; off > 0; off >>= 1)
      vmax[c] = fmaxf(vmax[c], __shfl_xor(vmax[c], off, 32));
  }
  __shared__ float red[8][25];
  const int lane = t & 31, wv = t >> 5;
  if (lane == 0) {
#pragma unroll
    for (int c = 0; c < 25; ++c) red[wv][c] = vmax[c];
  }
  __syncthreads();
  if (t < nch) {
    float v = -__builtin_inff();
#pragma unroll
    for (int w = 0; w < 8; ++w) v = fmaxf(v, red[w][t]);
    // values are either -inf (init) or positive -> signed-int max is exact
    atomicMax((int*)&ws[WS_M + (k * SS + cell) * 32 + t], __float_as_int(v));
  }
}

// ---------------- pass B: denom = sum exp(t_h - m_safe) * mask --------
__global__ __launch_bounds__(256)
void k_den(const float* __restrict__ y, const float* __restrict__ yh,
           float* __restrict__ ws) {
  const int t    = threadIdx.x;
  const int z    = blockIdx.x;
  const int cell = blockIdx.y;
  const int k    = blockIdx.z;
  const int nch  = 25 - 5 * k;
  const int cb   = 5 * k;

  const float* Mrow = ws + WS_M + (k * SS + cell) * 32;
  float msf[25];
#pragma unroll
  for (int c = 0; c < 25; ++c) {
    float m = (c < nch) ? Mrow[c] : 0.0f;
    msf[c] = (m == -__builtin_inff()) ? 0.0f : m;   // m_safe
  }
  float acc[25];
#pragma unroll
  for (int c = 0; c < 25; ++c) acc[c] = 0.0f;

  const size_t cellOff = (size_t)cell * CCH;
#pragma unroll
  for (int i = 0; i < 2; ++i) {
    int b = z * 512 + i * 256 + t;
    size_t rowo = (size_t)b * ROWF + cellOff;
    if (i == 0) __builtin_prefetch(&yh[rowo + (size_t)256 * ROWF], 0, 0);
    float conf = y[rowo + cb];
    if (conf == 1.0f) {
      const float* rh = yh + rowo + cb + 5;
#pragma unroll
      for (int c = 0; c < 25; ++c)
        if (c < nch) acc[c] += expf(rh[c] - msf[c]);
    }
  }
#pragma unroll
  for (int c = 0; c < 25; ++c) {
#pragma unroll
    for (int off = 16; off > 0; off >>= 1)
      acc[c] += __shfl_xor(acc[c], off, 32);
  }
  __shared__ float red[8][25];
  const int lane = t & 31, wv = t >> 5;
  if (lane == 0) {
#pragma unroll
    for (int c = 0; c < 25; ++c) red[wv][c] = acc[c];
  }
  __syncthreads();
  if (t < nch) {
    float v = 0.0f;
#pragma unroll
    for (int w = 0; w < 8; ++w) v += red[w][t];
    atomicAdd(&ws[WS_D + (k * SS + cell) * 32 + t], v);
  }
}

// ---------------- pass C: final loss (TDM + async-LDS staged) ---------
// 784 blocks x 256 threads; each thread owns one (b,cell) row of 30 ch.
// Block span in y / y_h is contiguous: 256 rows * 120 B = 30720 B each.
// y span: one Tensor-Data-Mover DMA (wave 0, TENSORcnt).
// y_h span: per-lane async b128 copies (all waves, ASYNCcnt).
__global__ __launch_bounds__(256)
void k_loss(const float* __restrict__ y, const float* __restrict__ yh,
            float* __restrict__ ws) {
  __shared__ __align__(16) float smem[15360];  // [0,7680): y, [7680,15360): y_h
  __shared__ float red2[8];
  const int t   = threadIdx.x;
  const int blk = blockIdx.x;
  const size_t baseF = (size_t)blk * 7680;     // floats

  const unsigned ldsY = lds_byte_off(&smem[0]);
  const unsigned ldsH = lds_byte_off(&smem[7680]);
  const char* gy = (const char*)(y + baseF);
  const char* gh = (const char*)(yh + baseF);

  // wave 0: single TDM descriptor moves the whole 30720-B y span
  if ((t >> 5) == 0) {
    tdm_load_span(ldsY, gy, 30720u);
  }
  // all waves: async per-lane b128 copies for the y_h span
  for (int c = t; c < 1920; c += 256) {        // 1920 x 16B chunks
    async_ld_b128(ldsH + c * 16, gh + (size_t)c * 16);
  }
  wait_async0();
  if ((t >> 5) == 0) {
    __builtin_amdgcn_s_wait_tensorcnt(0);
  }
  __syncthreads();

  const int p    = blk * 256 + t;
  const int cell = p % SS;

  float Y[30], H[30];
  const float2* py = (const float2*)&smem[t * 30];
  const float2* ph = (const float2*)&smem[7680 + t * 30];
#pragma unroll
  for (int j = 0; j < 15; ++j) {
    float2 a = py[j]; Y[2 * j] = a.x; Y[2 * j + 1] = a.y;
    float2 b = ph[j]; H[2 * j] = b.x; H[2 * j + 1] = b.y;
  }

  const float* M0 = ws + WS_M + cell * 32;
  const float* D0 = ws + WS_D + cell * 32;
  const float* M1 = ws + WS_M + (SS + cell) * 32;
  const float* D1 = ws + WS_D + (SS + cell) * 32;

  float total = 0.0f;

  // ---- box 0 ----
  {
    float d = Y[0] - H[0]; d *= d;
    const bool mk = (Y[0] == 1.0f);
    total += mk ? d : 0.5f * d;                 // obj + L_NOOBJ*noobj
    if (mk) {
      float dx = Y[1] - H[1], dy = Y[2] - H[2];
      float dw = sqrtf(Y[3]) - sqrtf(H[3]);
      float dh = sqrtf(Y[4]) - sqrtf(H[4]);
      total += 5.0f * (dx * dx + dy * dy + dw * dw + dh * dh);  // L_COORD
#pragma unroll
      for (int c = 0; c < 25; ++c) {            // tail = channels 5..29
        float m = M0[c];  m = (m == -__builtin_inff()) ? 0.0f : m;
        float den = D0[c]; den = (den > 0.0f) ? den : 1.0f;
        float sm = expf(H[5 + c] - m) / den;
        float e = Y[5 + c] - sm;
        total += e * e;
      }
    }
  }
  // ---- box 1 ----
  {
    float d = Y[5] - H[5]; d *= d;
    const bool mk = (Y[5] == 1.0f);
    total += mk ? d : 0.5f * d;
    if (mk) {
      float dx = Y[6] - H[6], dy = Y[7] - H[7];
      float dw = sqrtf(Y[8]) - sqrtf(H[8]);
      float dh = sqrtf(Y[9]) - sqrtf(H[9]);
      total += 5.0f * (dx * dx + dy * dy + dw * dw + dh * dh);
#pragma unroll
      for (int c = 0; c < 20; ++c) {            // tail = channels 10..29
        float m = M1[c];  m = (m == -__builtin_inff()) ? 0.0f : m;
        float den = D1[c]; den = (den > 0.0f) ? den : 1.0f;
        float sm = expf(H[10 + c] - m) / den;
        float e = Y[10 + c] - sm;
        total += e * e;
      }
    }
  }

  // block reduction: wave32 shfl then cross-wave LDS
#pragma unroll
  for (int off = 16; off > 0; off >>= 1) total += __shfl_xor(total, off, 32);
  if ((t & 31) == 0) red2[t >> 5] = total;
  __syncthreads();
  if (t == 0) {
    float s = 0.0f;
#pragma unroll
    for (int w = 0; w < 8; ++w) s += red2[w];
    atomicAdd(&ws[WS_ACC], s);
  }
}

// ---------------- final scale ----------------------------------------
__global__ void k_final(const float* __restrict__ ws, float* __restrict__ out) {
  if (threadIdx.x == 0 && blockIdx.x == 0)
    out[0] = ws[WS_ACC] * (1.0f / (float)NBATCH);
}

extern "C" void kernel_launch(void* const* d_in, const int* in_sizes, int n_in,
                              void* d_out, int out_size, void* d_ws, size_t ws_size,
                              hipStream_t stream) {
  (void)in_sizes; (void)n_in; (void)out_size; (void)ws_size;
  const float* y  = (const float*)d_in[0];
  const float* yh = (const float*)d_in[1];
  float* ws  = (float*)d_ws;
  float* out = (float*)d_out;

  k_init<<<25, 256, 0, stream>>>(ws);
  dim3 grid(8, SS, 2);
  k_max<<<grid, 256, 0, stream>>>(y, yh, ws);
  k_den<<<grid, 256, 0, stream>>>(y, yh, ws);
  k_loss<<<NPAIR / 256, 256, 0, stream>>>(y, yh, ws);
  k_final<<<1, 64, 0, stream>>>(ws, out);
}